// SoftMinMax_38817914422087
// MI455X (gfx1250) — compile-verified
//
#include <hip/hip_runtime.h>
#include <stdint.h>

typedef __attribute__((ext_vector_type(2))) float    v2f;
typedef __attribute__((ext_vector_type(8))) float    v8f;
typedef __attribute__((ext_vector_type(4))) uint32_t v4u;
typedef __attribute__((ext_vector_type(4))) int      v4i;
typedef __attribute__((ext_vector_type(8))) int      v8i;

#define SMM_EPS 1e-7f
#define SMM_LOG2E 1.4426950408889634f
#define SMM_CLIP2 43.280851226668914f   /* 30 * log2(e) */

// One wave = one batch row b and a tile of 16 units u.
// Per 4-d chunk, two V_WMMA_F32_16X16X4_F32 accumulate
//   Dnum[u,u'] = sum_d (x*e) * W[d,u'],  Dden[u,u'] = sum_d e * W[d,u']
// with e = exp2(clip(x * k[u]*log2e)); only the diagonal u'==u is kept.
// exp is computed once per (b,d,u); the 16x MAC redundancy rides the matrix
// pipe (kernel is transcendental-bound, HBM traffic ~6.5MB total).
__global__ __launch_bounds__(128) void softminmax_wmma_kernel(
    const float* __restrict__ xg, const float* __restrict__ Wg,
    const float* __restrict__ kg, float* __restrict__ out,
    int B, int D, int U)
{
    __shared__ float xs[512];      // one x row (D = 512)
    __shared__ float redN[1024];   // 4 waves * 32 lanes * 8 C-VGPRs
    __shared__ float redD[1024];

    const int tid   = threadIdx.x;
    const int wave  = tid >> 5;
    const int lane  = tid & 31;
    const int b     = blockIdx.x;
    const int uBase = blockIdx.y * 64 + wave * 16;
    const int uu    = uBase + (lane & 15);   // A row M / B col N
    const int kgrp  = lane >> 4;             // lanes 0-15: K0,1 ; 16-31: K2,3

    // ---- Stage x row via the Tensor Data Mover (wave 0) ------------------
    if (wave == 0) {
        const uint32_t ldsAddr = (uint32_t)(uintptr_t)(&xs[0]);
        const uint64_t ga = (uint64_t)(uintptr_t)(xg + (size_t)b * D);
        v4u g0;
        g0[0] = 1u;                                   // count=1 (valid D#)
        g0[1] = ldsAddr;                              // lds_addr [63:32]
        g0[2] = (uint32_t)(ga & 0xFFFFFFFFu);         // global_addr lo
        g0[3] = (uint32_t)((ga >> 32) & 0x01FFFFFFu)  // global_addr [56:32]
              | (2u << 30);                           // type=2 ("image")
        v8i g1;
        g1[0] = (int)(2u << 16);                      // data_size=2 (4B), no multicast
        g1[1] = (int)(((uint32_t)D & 0xFFFFu) << 16); // tensor_dim0 lo16
        g1[2] = (int)(1u << 16);                      // tensor_dim0 hi=0, tensor_dim1=1
        g1[3] = (int)(((uint32_t)D & 0xFFFFu) << 16); // tile_dim0 = D
        g1[4] = 1;                                    // tile_dim1 = 1
        g1[5] = D;                                    // tensor_dim0_stride = D
        g1[6] = (int)(((uint32_t)D & 0xFFFFu) << 16); // tensor_dim1_stride = D
        g1[7] = 0;
        v4i gz = {0, 0, 0, 0};
#if defined(__clang_major__) && (__clang_major__ >= 23)
        v8i gz8 = {0, 0, 0, 0, 0, 0, 0, 0};
        __builtin_amdgcn_tensor_load_to_lds(g0, g1, gz, gz, gz8, 0);
#else
        __builtin_amdgcn_tensor_load_to_lds(g0, g1, gz, gz, 0);
#endif
        __builtin_amdgcn_s_wait_tensorcnt(0);
    }
    // Redundant coalesced staging (writes identical values; keeps correctness
    // independent of descriptor details, costs one 2KB read from L2).
    for (int i = tid; i < D; i += 128) xs[i] = xg[(size_t)b * D + i];
    __syncthreads();

    // ---- Per-lane setup: 32-bit W offsets, pre-scaled exponent -----------
    const float ku2 = kg[uu] * SMM_LOG2E;
    uint32_t wOff = (uint32_t)uu + (uint32_t)(2 * kgrp) * (uint32_t)U;
    const uint32_t stepW = 16u * (uint32_t)U;
    uint32_t co[8];
#pragma unroll
    for (int c = 0; c < 4; ++c) {
        co[2 * c]     = (uint32_t)(4 * c)     * (uint32_t)U;
        co[2 * c + 1] = (uint32_t)(4 * c + 1) * (uint32_t)U;
    }

    // Software-pipelined W double buffer (hides L2 latency behind WMMA+exp).
    float wcur[8];
#pragma unroll
    for (int j = 0; j < 8; ++j) wcur[j] = Wg[wOff + co[j]];

    v8f nAcc[4] = {{}, {}, {}, {}};
    v8f dAcc[4] = {{}, {}, {}, {}};

    for (int d0 = 0; d0 < D; d0 += 16) {
        const uint32_t nOff = (d0 + 16 < D) ? (wOff + stepW) : wOff;
        float wnext[8];
#pragma unroll
        for (int j = 0; j < 8; ++j) wnext[j] = Wg[nOff + co[j]];

#pragma unroll
        for (int c = 0; c < 4; ++c) {
            const int dd = d0 + 4 * c + 2 * kgrp;
            const float x0 = xs[dd];
            const float x1 = xs[dd + 1];
            const float e0 =
                __builtin_amdgcn_exp2f(fminf(fmaxf(ku2 * x0, -SMM_CLIP2), SMM_CLIP2));
            const float e1 =
                __builtin_amdgcn_exp2f(fminf(fmaxf(ku2 * x1, -SMM_CLIP2), SMM_CLIP2));
            v2f bm = {wcur[2 * c], wcur[2 * c + 1]};  // B[k,u'] = W[d,u']
            v2f ad = {e0, e1};                        // A[u,k]  = e
            v2f an = {x0 * e0, x1 * e1};              // A[u,k]  = x*e
            nAcc[c] = __builtin_amdgcn_wmma_f32_16x16x4_f32(
                          false, an, false, bm, (short)0, nAcc[c], false, false);
            dAcc[c] = __builtin_amdgcn_wmma_f32_16x16x4_f32(
                          false, ad, false, bm, (short)0, dAcc[c], false, false);
        }

#pragma unroll
        for (int j = 0; j < 8; ++j) wcur[j] = wnext[j];
        wOff = nOff;
    }

    v8f sN = (nAcc[0] + nAcc[1]) + (nAcc[2] + nAcc[3]);
    v8f sD = (dAcc[0] + dAcc[1]) + (dAcc[2] + dAcc[3]);

    // Spill the 8 C/D VGPRs so any lane can read the diagonal element.
    const int base = (wave * 32 + lane) * 8;
#pragma unroll
    for (int r = 0; r < 8; ++r) { redN[base + r] = sN[r]; redD[base + r] = sD[r]; }
    __syncthreads();

    // Diagonal (u,u): u<8 -> lane u, VGPR u ; u>=8 -> lane u+16, VGPR u-8.
    if (lane < 16) {
        const int L   = (lane < 8) ? lane : lane + 16;
        const int r   = lane & 7;
        const int idx = (wave * 32 + L) * 8 + r;
        const float num = redN[idx];
        const float den = redD[idx] + SMM_EPS;
        out[(size_t)b * U + uBase + lane] = num / den;
    }
}

extern "C" void kernel_launch(void* const* d_in, const int* in_sizes, int n_in,
                              void* d_out, int out_size, void* d_ws, size_t ws_size,
                              hipStream_t stream) {
    const float* x = (const float*)d_in[0];   // [B, D]
    const float* W = (const float*)d_in[1];   // [D, U]
    const float* k = (const float*)d_in[2];   // [U]
    float* out = (float*)d_out;               // [B, U]

    const int U = in_sizes[2];
    const int D = (U > 0) ? in_sizes[1] / U : 0;
    const int B = (D > 0) ? in_sizes[0] / D : 0;

    dim3 grid(B, U / 64);   // B=2048, U/64=4 -> 8192 blocks, 32768 waves
    dim3 block(128);        // 4 wave32 waves per block
    softminmax_wmma_kernel<<<grid, block, 0, stream>>>(x, W, k, out, B, D, U);
}